// PtGriddingLoss_72301479461303
// MI455X (gfx1250) — compile-verified
//
#include <hip/hip_runtime.h>
#include <hip/hip_bf16.h>

typedef __attribute__((ext_vector_type(2))) float v2f;
typedef __attribute__((ext_vector_type(8))) float v8f;

#define SENT 1.0e6f

// ---------------------------------------------------------------------------
// Kernel 0: zero the scalar output
// ---------------------------------------------------------------------------
__global__ void zero_out_kernel(float* out) {
    if (threadIdx.x == 0 && blockIdx.x == 0) out[0] = 0.0f;
}

// ---------------------------------------------------------------------------
// Kernel 1: unproject depth -> pred xyz, validity, and pack each side twice:
//   qpack (query/A side):  (x, y, z, 1)          -- pure b64 A-operand loads
//   tpack (target/B side): (-2x, -2y, -2z, |p|^2) -- pure b64 B-operand loads
//   pp:    |p|^2                                  -- C-operand gather
// Invalid points get far sentinels: P side +S, G side -S (so cross distances
// are ~S^2 and never win a min; masked sums drop them entirely).
// ---------------------------------------------------------------------------
__global__ void precompute_kernel(const float* __restrict__ pred,
                                  const float* __restrict__ gt,
                                  const int*   __restrict__ mask,
                                  const float* __restrict__ pfx,
                                  const float* __restrict__ pfy,
                                  const float* __restrict__ pcx,
                                  const float* __restrict__ pcy,
                                  float* __restrict__ qpackP,
                                  float* __restrict__ tpackP,
                                  float* __restrict__ ppP,
                                  float* __restrict__ qpackG,
                                  float* __restrict__ tpackG,
                                  float* __restrict__ ppG,
                                  int*   __restrict__ vp,
                                  int*   __restrict__ vg,
                                  int B, int H, int W) {
    int tid = blockIdx.x * blockDim.x + threadIdx.x;
    int N = H * W;
    if (tid >= B * N) return;
    int b = tid / N;
    int i = tid - b * N;
    int v = i / W;
    int u = i - v * W;

    float fx = *pfx, fy = *pfy, cx = *pcx, cy = *pcy;

    float z = pred[b * N + i];
    float x = ((float)u - cx) / fx * z;
    float y = ((float)v - cy) / fy * z;

    int m = mask[b * N + i] > 0;
    int pvalid = m && ((x + y + z) != 0.0f);

    float gx = gt[(b * 3 + 0) * N + i];
    float gy = gt[(b * 3 + 1) * N + i];
    float gz = gt[(b * 3 + 2) * N + i];
    int gvalid = m && ((gx + gy + gz) != 0.0f);

    float px = pvalid ? x : SENT;
    float py = pvalid ? y : 0.0f;
    float pz = pvalid ? z : 0.0f;
    float pp = px * px + py * py + pz * pz;
    ((float4*)qpackP)[tid] = make_float4(px, py, pz, 1.0f);
    ((float4*)tpackP)[tid] = make_float4(-2.0f * px, -2.0f * py, -2.0f * pz, pp);
    ppP[tid] = pp;

    float qx = gvalid ? gx : -SENT;
    float qy = gvalid ? gy : 0.0f;
    float qz = gvalid ? gz : 0.0f;
    float qq = qx * qx + qy * qy + qz * qz;
    ((float4*)qpackG)[tid] = make_float4(qx, qy, qz, 1.0f);
    ((float4*)tpackG)[tid] = make_float4(-2.0f * qx, -2.0f * qy, -2.0f * qz, qq);
    ppG[tid] = qq;

    vp[tid] = pvalid;
    vg[tid] = gvalid;
}

// ---------------------------------------------------------------------------
// Kernel 2: chamfer row-min via V_WMMA_F32_16X16X4_F32.
// One wave owns TWO 16-row tiles (32 query rows); each B tile load (256B,
// fully coalesced, L2-hot) feeds two WMMAs. Unroll-4 over B tiles gives a
// 4-deep s_clause of global_load_b64 and 8 WMMAs of independent work to hide
// load latency at low per-SIMD occupancy. blockIdx.y selects direction:
//   y=0: rows = pred, cols = gt   -> rowminP (d1)
//   y=1: rows = gt,   cols = pred -> rowminG (d2)
// d[i][j] = A(i,:).B(:,j) + C = (-2 q.t + |t|^2) + |q|^2 ; clamp applied
// once after the min-reduction (max(0,.) is monotone, commutes with min).
//   A layout (16x4 f32): lane = M (lane&15); half = lane>>4 -> K {0,1}/{2,3}
//   B layout (4x16 f32): lane = N (lane&15); half -> K base; VGPR = K in half
//   C/D layout: VGPR k -> M = k + half*8; N = lane&15
// ---------------------------------------------------------------------------
__global__ void __launch_bounds__(256)
chamfer_min_kernel(const float* __restrict__ qpackP,
                   const float* __restrict__ ppP,
                   const float* __restrict__ tpackP,
                   const float* __restrict__ qpackG,
                   const float* __restrict__ ppG,
                   const float* __restrict__ tpackG,
                   float* __restrict__ rowminP,
                   float* __restrict__ rowminG,
                   int N) {
    int dir = blockIdx.y;
    const float* qpack = dir ? qpackG : qpackP;
    const float* ppq   = dir ? ppG    : ppP;
    const float* tpack = dir ? tpackP : tpackG;   // target = the other side
    float* rowmin      = dir ? rowminG : rowminP;

    int gtid = blockIdx.x * blockDim.x + threadIdx.x;
    int wave = gtid >> 5;
    int lane = threadIdx.x & 31;
    int wavesPerBatch = N >> 5;              // 32 rows per wave
    int b  = wave / wavesPerBatch;
    int wt = wave - b * wavesPerBatch;
    int rowbase = wt * 32;

    int half = (lane >> 4) & 1;
    int mrow = lane & 15;
    int col  = lane & 15;

    // --- two A tiles: (x,y,z,1) homogeneous query rows, pure b64 loads ---
    v2f a0 = *(const v2f*)(qpack + ((size_t)(b * N + rowbase + mrow)) * 4 + half * 2);
    v2f a1 = *(const v2f*)(qpack + ((size_t)(b * N + rowbase + 16 + mrow)) * 4 + half * 2);

    // --- two C tiles: |q|^2 broadcast along rows ---
    v8f c0, c1;
#pragma unroll
    for (int k = 0; k < 8; ++k) {
        c0[k] = ppq[(size_t)b * N + rowbase + k + half * 8];
        c1[k] = ppq[(size_t)b * N + rowbase + 16 + k + half * 8];
    }

    float mins0[8], mins1[8];
#pragma unroll
    for (int k = 0; k < 8; ++k) { mins0[k] = 3.0e38f; mins1[k] = 3.0e38f; }

    const float* bptr = tpack + (size_t)b * N * 4 + col * 4 + half * 2;
    int tiles = N >> 4;                      // 1024, divisible by 4

#pragma unroll 4
    for (int jt = 0; jt < tiles; ++jt) {
        v2f bv = *(const v2f*)bptr;          // (-2x,-2y) or (-2z,|t|^2): zero VALU prep
        bptr += 64;                          // next 16 target points

        v8f d0 = __builtin_amdgcn_wmma_f32_16x16x4_f32(
            false, a0, false, bv, (short)0, c0, false, false);
#pragma unroll
        for (int k = 0; k < 8; ++k) mins0[k] = fminf(mins0[k], d0[k]);

        v8f d1 = __builtin_amdgcn_wmma_f32_16x16x4_f32(
            false, a1, false, bv, (short)0, c1, false, false);
#pragma unroll
        for (int k = 0; k < 8; ++k) mins1[k] = fminf(mins1[k], d1[k]);
    }

    // reduce across the 16 lanes of each half (xor<16 stays within the half)
#pragma unroll
    for (int m = 1; m <= 8; m <<= 1) {
#pragma unroll
        for (int k = 0; k < 8; ++k) {
            mins0[k] = fminf(mins0[k], __shfl_xor(mins0[k], m, 32));
            mins1[k] = fminf(mins1[k], __shfl_xor(mins1[k], m, 32));
        }
    }

    if ((lane & 15) == 0) {
#pragma unroll
        for (int k = 0; k < 8; ++k) {
            rowmin[(size_t)b * N + rowbase + k + half * 8]      = fmaxf(mins0[k], 0.0f);
            rowmin[(size_t)b * N + rowbase + 16 + k + half * 8] = fmaxf(mins1[k], 0.0f);
        }
    }
}

// ---------------------------------------------------------------------------
// Kernel 3: per-batch masked sums + has_p/has_g gating, atomicAdd(out, s/B)
// ---------------------------------------------------------------------------
__global__ void chamfer_reduce_kernel(const float* __restrict__ rowminP,
                                      const float* __restrict__ rowminG,
                                      const int* __restrict__ vp,
                                      const int* __restrict__ vg,
                                      float* __restrict__ out,
                                      int N, float invB) {
    int b = blockIdx.x;
    int tid = threadIdx.x;
    float s1 = 0.0f, s2 = 0.0f;
    int aP = 0, aG = 0;
    for (int i = tid; i < N; i += blockDim.x) {
        int p = vp[b * N + i];
        int g = vg[b * N + i];
        aP |= p; aG |= g;
        if (p) s1 += rowminP[b * N + i];
        if (g) s2 += rowminG[b * N + i];
    }
    __shared__ float sh1[256];
    __shared__ float sh2[256];
    __shared__ int shp, shg;
    if (tid == 0) { shp = 0; shg = 0; }
    __syncthreads();
    sh1[tid] = s1; sh2[tid] = s2;
    if (aP) atomicOr(&shp, 1);
    if (aG) atomicOr(&shg, 1);
    __syncthreads();
    for (int off = 128; off > 0; off >>= 1) {
        if (tid < off) {
            sh1[tid] += sh1[tid + off];
            sh2[tid] += sh2[tid + off];
        }
        __syncthreads();
    }
    if (tid == 0) {
        float tot = (shg ? sh1[0] : 0.0f) + (shp ? sh2[0] : 0.0f);
        atomicAdd(out, tot * invB);
    }
}

// ---------------------------------------------------------------------------
extern "C" void kernel_launch(void* const* d_in, const int* in_sizes, int n_in,
                              void* d_out, int out_size, void* d_ws, size_t ws_size,
                              hipStream_t stream) {
    const int H = 128, W = 128;
    const int N = H * W;
    const int B = in_sizes[0] / N;   // pred is (B,1,H,W)

    const float* pred = (const float*)d_in[0];
    const float* gt   = (const float*)d_in[1];
    const int*   mask = (const int*)d_in[2];
    const float* pfx  = (const float*)d_in[3];
    const float* pfy  = (const float*)d_in[4];
    const float* pcx  = (const float*)d_in[5];
    const float* pcy  = (const float*)d_in[6];
    float* out = (float*)d_out;

    // workspace layout (floats)
    size_t BN = (size_t)B * N;
    float* qpackP  = (float*)d_ws;          // BN*4
    float* tpackP  = qpackP + BN * 4;       // BN*4
    float* ppP     = tpackP + BN * 4;       // BN
    float* qpackG  = ppP + BN;              // BN*4
    float* tpackG  = qpackG + BN * 4;       // BN*4
    float* ppG     = tpackG + BN * 4;       // BN
    float* rowminP = ppG + BN;              // BN
    float* rowminG = rowminP + BN;          // BN
    int*   vp      = (int*)(rowminG + BN);  // BN
    int*   vg      = vp + BN;               // BN

    zero_out_kernel<<<1, 64, 0, stream>>>(out);

    {
        int total = B * N;
        int threads = 256;
        int blocks = (total + threads - 1) / threads;
        precompute_kernel<<<blocks, threads, 0, stream>>>(
            pred, gt, mask, pfx, pfy, pcx, pcy,
            qpackP, tpackP, ppP, qpackG, tpackG, ppG, vp, vg, B, H, W);
    }

    {
        int waves = B * (N / 32);            // one wave per 32-row strip
        int threads = 256;                   // 8 wave32 per block
        dim3 grid((waves * 32 + threads - 1) / threads, 2);  // y = direction
        chamfer_min_kernel<<<grid, threads, 0, stream>>>(
            qpackP, ppP, tpackP, qpackG, ppG, tpackG, rowminP, rowminG, N);
    }

    chamfer_reduce_kernel<<<B, 256, 0, stream>>>(rowminP, rowminG, vp, vg,
                                                 out, N, 1.0f / (float)B);
}